// FSSB_80676665688846
// MI455X (gfx1250) — compile-verified
//
#include <hip/hip_runtime.h>
#include <math.h>

#define B_  2
#define H_  64
#define W__ 64
#define C_  96
#define L_  4096
#define DI_ 192
#define N_  16
#define R_  6
#define K_  4

typedef __attribute__((ext_vector_type(2))) float v2f;
typedef __attribute__((ext_vector_type(8))) float v8f;

__device__ __forceinline__ float silu_f(float x){ return x / (1.0f + expf(-x)); }
__device__ __forceinline__ float softplus_f(float x){ return fmaxf(x, 0.0f) + log1pf(expf(-fabsf(x))); }
__device__ __forceinline__ float gelu_f(float x){ return 0.5f * x * (1.0f + erff(x * 0.70710678118654752f)); }

// ---------------------------------------------------------------------------
// Full-tile strided batched GEMM using V_WMMA_F32_16X16X4_F32.
// Requirements (guaranteed by caller): M,N multiples of 16; K multiple of 8.
// C(m,n) = sum_k A(m,k)*B(k,n);  A(m,k)=A[m*sA0+k*sA1], etc.
// AK1/BK1: K-stride of A/B is 1 -> vector (b64) loads for the lane's K-pair.
// epi: 0=none, 1=+bias[n], 2=+bias[n] then SiLU
// One wave computes one 16x16 C tile; 2 WMMAs (K=8) per loop iteration.
// ---------------------------------------------------------------------------
template<bool AK1, bool BK1>
__global__ __launch_bounds__(32)
void k_gemm_f(const float* __restrict__ A, const float* __restrict__ Bm,
              float* __restrict__ Cm, const float* __restrict__ bias,
              int K,
              long long sA0, long long sA1,
              long long sB0, long long sB1,
              long long sC0, long long sC1,
              long long bA, long long bB, long long bC, int epi)
{
    int z = blockIdx.z;
    int tm = blockIdx.x * 16, tn = blockIdx.y * 16;
    int lane = threadIdx.x;
    int half = lane >> 4, l16 = lane & 15;
    int row = tm + l16;   // A row handled by this lane
    int col = tn + l16;   // B col handled by this lane

    // lanes 0-15 hold K = k0, k0+1 ; lanes 16-31 hold K = k0+2, k0+3
    const float* ap = A  + (long long)z * bA + (long long)row * sA0 + (long long)(half * 2) * sA1;
    const float* bp = Bm + (long long)z * bB + (long long)(half * 2) * sB0 + (long long)col * sB1;
    const long long stA4 = 4 * sA1, stB4 = 4 * sB0;
    const long long stA8 = 8 * sA1, stB8 = 8 * sB0;

    v8f acc = {0.f,0.f,0.f,0.f,0.f,0.f,0.f,0.f};
    for (int k0 = 0; k0 < K; k0 += 8) {
        v2f a0, a1, b0, b1;
        if (AK1) { a0 = *(const v2f*)ap; a1 = *(const v2f*)(ap + 4); }
        else     { a0.x = ap[0]; a0.y = ap[sA1]; a1.x = ap[stA4]; a1.y = ap[stA4 + sA1]; }
        if (BK1) { b0 = *(const v2f*)bp; b1 = *(const v2f*)(bp + 4); }
        else     { b0.x = bp[0]; b0.y = bp[sB0]; b1.x = bp[stB4]; b1.y = bp[stB4 + sB0]; }
        acc = __builtin_amdgcn_wmma_f32_16x16x4_f32(false, a0, false, b0, (short)0, acc, false, false);
        acc = __builtin_amdgcn_wmma_f32_16x16x4_f32(false, a1, false, b1, (short)0, acc, false, false);
        ap += stA8; bp += stB8;
    }

    float bb = (epi >= 1) ? bias[col] : 0.0f;
    float* cp = Cm + (long long)z * bC + (long long)(tm + half * 8) * sC0 + (long long)col * sC1;
    #pragma unroll
    for (int r = 0; r < 8; ++r) {     // VGPR r: lanes0-15 -> M=tm+r, lanes16-31 -> M=tm+8+r
        float v = acc[r] + bb;
        if (epi == 2) v = silu_f(v);
        cp[(long long)r * sC0] = v;
    }
}

// Pad x_proj_w (4,38,192) -> (4,48,192) with zero rows so x_proj GEMM is full-tile.
__global__ __launch_bounds__(256)
void k_pad_xpw(const float* __restrict__ w, float* __restrict__ wp, int total)
{
    int idx = blockIdx.x * blockDim.x + threadIdx.x;
    if (idx >= total) return;
    int col = idx % DI_;
    int r = (idx / DI_) % 48;
    int k = idx / (DI_ * 48);
    wp[idx] = (r < 38) ? w[((long long)k * 38 + r) * DI_ + col] : 0.0f;
}

// ---------------------------------------------------------------------------
// LayerNorm over last dim C_=96, one wave32 per row.
// ---------------------------------------------------------------------------
__global__ __launch_bounds__(256)
void k_ln96(const float* __restrict__ x, const float* __restrict__ g,
            const float* __restrict__ bvec, float* __restrict__ out,
            int rows, float eps)
{
    int warp = (blockIdx.x * blockDim.x + threadIdx.x) >> 5;
    int lane = threadIdx.x & 31;
    if (warp >= rows) return;
    const float* r = x + (long long)warp * C_;
    float v0 = r[lane], v1 = r[lane + 32], v2 = r[lane + 64];
    float s = v0 + v1 + v2;
    #pragma unroll
    for (int o = 16; o > 0; o >>= 1) s += __shfl_xor(s, o, 32);
    float mean = s * (1.0f / C_);
    float d0 = v0 - mean, d1 = v1 - mean, d2 = v2 - mean;
    float q = d0*d0 + d1*d1 + d2*d2;
    #pragma unroll
    for (int o = 16; o > 0; o >>= 1) q += __shfl_xor(q, o, 32);
    float inv = rsqrtf(q * (1.0f / C_) + eps);
    float* ro = out + (long long)warp * C_;
    ro[lane]      = d0 * inv * g[lane]      + bvec[lane];
    ro[lane + 32] = d1 * inv * g[lane + 32] + bvec[lane + 32];
    ro[lane + 64] = d2 * inv * g[lane + 64] + bvec[lane + 64];
}

// ---------------------------------------------------------------------------
// Depthwise conv KSxKS, SAME. Input indexed as in[b*inB + c*inC + (h*64+w)*inP].
// Output channel-plane layout (B, Cn, 64, 64) contiguous.
// ---------------------------------------------------------------------------
template<int KS>
__global__ __launch_bounds__(256)
void k_dwconv(const float* __restrict__ in, long long inB, long long inC, long long inP,
              const float* __restrict__ w, const float* __restrict__ bias,
              float* __restrict__ out, int Cn, int act, int total)
{
    int idx = blockIdx.x * blockDim.x + threadIdx.x;
    if (idx >= total) return;
    int hw = idx & (L_ - 1);
    int bc = idx >> 12;
    int c = bc % Cn, b = bc / Cn;
    int h = hw >> 6, ww = hw & 63;
    const float* ip = in + (long long)b * inB + (long long)c * inC;
    const float* wp = w + c * KS * KS;
    float acc = 0.0f;
    #pragma unroll
    for (int kh = 0; kh < KS; ++kh) {
        int ih = h + kh - KS / 2;
        if (ih < 0 || ih >= H_) continue;
        #pragma unroll
        for (int kw = 0; kw < KS; ++kw) {
            int iw = ww + kw - KS / 2;
            if (iw < 0 || iw >= W__) continue;
            acc += ip[(long long)(ih * W__ + iw) * inP] * wp[kh * KS + kw];
        }
    }
    if (bias) acc += bias[c];
    if (act) acc = silu_f(acc);
    out[idx] = acc;
}

// ---------------------------------------------------------------------------
// Build 4-direction xs (B,4,DI,L) from xc channel planes (B,DI,64,64).
// ---------------------------------------------------------------------------
__global__ __launch_bounds__(256)
void k_build_xs(const float* __restrict__ xc, float* __restrict__ xs, int total)
{
    int idx = blockIdx.x * blockDim.x + threadIdx.x;
    if (idx >= total) return;
    int l = idx & (L_ - 1);
    int bd = idx >> 12;
    int d = bd % DI_, b = bd / DI_;
    int h = l >> 6, w = l & 63;
    int lT = w * 64 + h;
    float v = xc[idx];
    long long base = (((long long)b * 4) * DI_ + d) * L_;
    xs[base + l] = v;
    xs[base + (long long)DI_ * L_ + lT] = v;
    xs[base + 2LL * DI_ * L_ + (L_ - 1 - l)] = v;
    xs[base + 3LL * DI_ * L_ + (L_ - 1 - lT)] = v;
}

// ---------------------------------------------------------------------------
// Selective scan. One block per (b,k); lane d owns one chain with h[16] in regs.
// Per-step shared dts(6)/Bs(16)/Cs(16) staged in LDS, 16 steps per barrier pair.
// dt-projection + softplus fused in. Writes out_y = y + u*D.
// xdbl is padded to 48 channels per (b,k); only first 38 are live.
// ---------------------------------------------------------------------------
__global__ __launch_bounds__(192)
void k_scan(const float* __restrict__ xs, const float* __restrict__ xdbl,
            const float* __restrict__ Alogs, const float* __restrict__ dtw,
            const float* __restrict__ dtb, const float* __restrict__ Dsp,
            float* __restrict__ oy)
{
    const int CH = 16;
    const int CC = R_ + 2 * N_;            // 38 shared channels per step
    int bk = blockIdx.x;                   // b*4+k
    int k = bk & 3;
    int d = threadIdx.x;
    int kd = k * DI_ + d;
    const float* xrow = xs + ((long long)bk * DI_ + d) * L_;
    const float* xd   = xdbl + (long long)bk * 48 * L_;   // padded stride
    float* orow = oy + ((long long)bk * DI_ + d) * L_;

    float Av[N_], hst[N_];
    #pragma unroll
    for (int n = 0; n < N_; ++n) { Av[n] = -expf(Alogs[kd * N_ + n]); hst[n] = 0.0f; }
    float wv[R_];
    #pragma unroll
    for (int r = 0; r < R_; ++r) wv[r] = dtw[kd * R_ + r];
    float dtbv = dtb[kd];
    float Dv = Dsp[kd];

    __shared__ float sh[CC * CH];
    for (int l0 = 0; l0 < L_; l0 += CH) {
        for (int t = threadIdx.x; t < CC * CH; t += blockDim.x) {
            int c = t / CH, s2 = t - c * CH;
            sh[s2 * CC + c] = xd[(long long)c * L_ + l0 + s2];
        }
        if (l0 + CH < L_) __builtin_prefetch(xrow + l0 + CH, 0, 1);
        __syncthreads();
        #pragma unroll 1
        for (int s2 = 0; s2 < CH; ++s2) {
            const float* row = sh + s2 * CC;
            float dtv = dtbv;
            #pragma unroll
            for (int r = 0; r < R_; ++r) dtv += wv[r] * row[r];
            dtv = softplus_f(dtv);
            float u = xrow[l0 + s2];
            float du = dtv * u;
            float y = 0.0f;
            #pragma unroll
            for (int n = 0; n < N_; ++n) {
                float dA = expf(dtv * Av[n]);
                hst[n] = hst[n] * dA + du * row[R_ + n];
                y += hst[n] * row[R_ + N_ + n];
            }
            orow[l0 + s2] = y + u * Dv;
        }
        __syncthreads();
    }
}

// ---------------------------------------------------------------------------
// Merge 4 directions + out-norm LN(192) + SiLU(z) gate. One block per (b,l).
// ---------------------------------------------------------------------------
__global__ __launch_bounds__(192)
void k_merge(const float* __restrict__ oy, const float* __restrict__ xz,
             const float* __restrict__ g, const float* __restrict__ be,
             float* __restrict__ ymul)
{
    int bl = blockIdx.x;
    int b = bl >> 12, l = bl & (L_ - 1);
    int h = l >> 6, w = l & 63;
    int lT = w * 64 + h;
    int d = threadIdx.x;
    long long base = (long long)b * 4 * DI_ * L_;
    float s = oy[base + ((long long)0 * DI_ + d) * L_ + l]
            + oy[base + ((long long)2 * DI_ + d) * L_ + (L_ - 1 - l)]
            + oy[base + ((long long)1 * DI_ + d) * L_ + lT]
            + oy[base + ((long long)3 * DI_ + d) * L_ + (L_ - 1 - lT)];
    __shared__ float red[256];
    red[d] = s; if (d < 64) red[DI_ + d] = 0.0f;
    __syncthreads();
    for (int st2 = 128; st2 > 0; st2 >>= 1) { if (d < st2) red[d] += red[d + st2]; __syncthreads(); }
    float mean = red[0] * (1.0f / DI_);
    __syncthreads();
    float dm = s - mean;
    red[d] = dm * dm; if (d < 64) red[DI_ + d] = 0.0f;
    __syncthreads();
    for (int st2 = 128; st2 > 0; st2 >>= 1) { if (d < st2) red[d] += red[d + st2]; __syncthreads(); }
    float inv = rsqrtf(red[0] * (1.0f / DI_) + 1e-5f);
    float z = xz[(long long)bl * (2 * DI_) + DI_ + d];
    ymul[(long long)bl * DI_ + d] = (dm * inv * g[d] + be[d]) * silu_f(z);
}

// ---------------------------------------------------------------------------
// 64-point dense DFT stages (ortho => 1/8 scale per stage), LDS twiddle tables.
// ---------------------------------------------------------------------------
__global__ __launch_bounds__(256)
void k_dft_r2c_w(const float* __restrict__ x, float* __restrict__ Fre,
                 float* __restrict__ Fim, int total)
{
    __shared__ float ct[64], st[64];
    int tid = threadIdx.x;
    if (tid < 64) { float a = (float)tid * (6.28318530717958647692f / 64.0f); ct[tid] = cosf(a); st[tid] = sinf(a); }
    __syncthreads();
    int idx = blockIdx.x * blockDim.x + tid;
    if (idx >= total) return;
    int wf = idx % 33; int t = idx / 33; int h = t & 63; int bc = t >> 6;
    const float* row = x + ((long long)bc * 64 + h) * 64;
    float re = 0.0f, im = 0.0f;
    for (int w = 0; w < 64; ++w) {
        int a = (w * wf) & 63;
        float v = row[w];
        re += v * ct[a];
        im -= v * st[a];
    }
    Fre[idx] = re * 0.125f; Fim[idx] = im * 0.125f;
}

__global__ __launch_bounds__(256)
void k_dft_h_fwd(const float* __restrict__ Fre, const float* __restrict__ Fim,
                 float* __restrict__ fr, int total)
{
    __shared__ float ct[64], st[64];
    int tid = threadIdx.x;
    if (tid < 64) { float a = (float)tid * (6.28318530717958647692f / 64.0f); ct[tid] = cosf(a); st[tid] = sinf(a); }
    __syncthreads();
    int idx = blockIdx.x * blockDim.x + tid;
    if (idx >= total) return;
    int wf = idx % 33; int t = idx / 33; int hf = t & 63; int bc = t >> 6;
    int b = bc / 48, c = bc % 48;
    long long base = ((long long)bc * 64) * 33 + wf;
    float re = 0.0f, im = 0.0f;
    for (int h = 0; h < 64; ++h) {
        int a = (h * hf) & 63;
        float xr = Fre[base + h * 33], xi = Fim[base + h * 33];
        re += xr * ct[a] + xi * st[a];   // *(cos - i sin)
        im += xi * ct[a] - xr * st[a];
    }
    long long ob = (((long long)b * 96 + 2 * c) * 64 + hf) * 33 + wf;
    fr[ob] = re * 0.125f;
    fr[ob + 64 * 33] = im * 0.125f;
}

__global__ __launch_bounds__(256)
void k_dft_h_inv(const float* __restrict__ frm, float* __restrict__ IHre,
                 float* __restrict__ IHim, int total)
{
    __shared__ float ct[64], st[64];
    int tid = threadIdx.x;
    if (tid < 64) { float a = (float)tid * (6.28318530717958647692f / 64.0f); ct[tid] = cosf(a); st[tid] = sinf(a); }
    __syncthreads();
    int idx = blockIdx.x * blockDim.x + tid;
    if (idx >= total) return;
    int wf = idx % 33; int t = idx / 33; int hf = t & 63; int bc = t >> 6;
    int b = bc / 48, c = bc % 48;
    long long ibr = (((long long)b * 96 + 2 * c) * 64) * 33 + wf;
    long long ibi = ibr + 64 * 33;
    float re = 0.0f, im = 0.0f;
    for (int h = 0; h < 64; ++h) {
        int a = (h * hf) & 63;
        float xr = frm[ibr + h * 33], xi = frm[ibi + h * 33];
        re += xr * ct[a] - xi * st[a];   // *(cos + i sin)
        im += xi * ct[a] + xr * st[a];
    }
    IHre[idx] = re * 0.125f; IHim[idx] = im * 0.125f;
}

__global__ __launch_bounds__(256)
void k_idft_c2r_w(const float* __restrict__ IHre, const float* __restrict__ IHim,
                  float* __restrict__ xo, int total)
{
    __shared__ float ct[64], st[64];
    int tid = threadIdx.x;
    if (tid < 64) { float a = (float)tid * (6.28318530717958647692f / 64.0f); ct[tid] = cosf(a); st[tid] = sinf(a); }
    __syncthreads();
    int idx = blockIdx.x * blockDim.x + tid;
    if (idx >= total) return;
    int w = idx & 63; int t = idx >> 6; int h = t & 63; int bc = t >> 6;
    long long base = ((long long)bc * 64 + h) * 33;
    float acc = 0.0f;
    for (int k2 = 0; k2 < 64; ++k2) {
        int kk = (k2 <= 32) ? k2 : 64 - k2;
        float sg = (k2 <= 32) ? 1.0f : -1.0f;
        float xr = IHre[base + kk], xi = sg * IHim[base + kk];
        int a = (w * k2) & 63;
        acc += xr * ct[a] - xi * st[a];
    }
    xo[idx] = acc * 0.125f;
}

// ---------------------------------------------------------------------------
// Elementwise / small kernels
// ---------------------------------------------------------------------------
__global__ __launch_bounds__(256)
void k_combine(const float* __restrict__ inx, const float* __restrict__ skip,
               const float* __restrict__ attn, const float* __restrict__ fft,
               float* __restrict__ xres, int total)
{
    int idx = blockIdx.x * blockDim.x + threadIdx.x;
    if (idx >= total) return;
    int c = idx % C_;
    xres[idx] = inx[idx] * skip[c] + attn[idx] + fft[idx];
}

__global__ __launch_bounds__(256)
void k_gelu_mul(const float* __restrict__ d1, const float* __restrict__ x2d,
                float* __restrict__ ym, int total)
{
    int idx = blockIdx.x * blockDim.x + threadIdx.x;
    if (idx >= total) return;
    int l = idx & (L_ - 1);
    int bc = idx >> 12;
    int c = bc % DI_, b = bc / DI_;
    float a = d1[((long long)b * 288 + c) * L_ + l];
    float g = (c < 96) ? d1[((long long)b * 288 + 192 + c) * L_ + l]
                       : x2d[((long long)b * 96 + (c - 96)) * L_ + l];
    ym[idx] = gelu_f(a) * g;
}

__global__ __launch_bounds__(256)
void k_pool(const float* __restrict__ y2, float* __restrict__ pool)
{
    int bc = blockIdx.x;   // b*96+c
    __shared__ float red[256];
    float s = 0.0f;
    for (int i = threadIdx.x; i < L_; i += 256) s += y2[(long long)bc * L_ + i];
    red[threadIdx.x] = s; __syncthreads();
    for (int st2 = 128; st2 > 0; st2 >>= 1) { if (threadIdx.x < st2) red[threadIdx.x] += red[threadIdx.x + st2]; __syncthreads(); }
    if (threadIdx.x == 0) pool[bc] = red[0] * (1.0f / L_);
}

__global__ __launch_bounds__(96)
void k_ca(const float* __restrict__ pool, const float* __restrict__ w1,
          const float* __restrict__ b1, const float* __restrict__ w2,
          const float* __restrict__ b2, float* __restrict__ att)
{
    int b = blockIdx.x; int o = threadIdx.x;
    __shared__ float t[3];
    if (o < 3) {
        float s = b1[o];
        for (int i = 0; i < 96; ++i) s += pool[b * 96 + i] * w1[o * 96 + i];
        t[o] = fmaxf(s, 0.0f);
    }
    __syncthreads();
    float s = b2[o];
    #pragma unroll
    for (int j = 0; j < 3; ++j) s += t[j] * w2[o * 3 + j];
    att[b * 96 + o] = 1.0f / (1.0f + expf(-s));
}

__global__ __launch_bounds__(256)
void k_final(const float* __restrict__ xres, const float* __restrict__ skip2,
             const float* __restrict__ y2, const float* __restrict__ att,
             float* __restrict__ outp, int total)
{
    int idx = blockIdx.x * blockDim.x + threadIdx.x;
    if (idx >= total) return;
    int c = idx % C_;
    int bl = idx / C_;
    int l = bl & (L_ - 1); int b = bl >> 12;
    outp[idx] = xres[idx] * skip2[c] + y2[((long long)b * C_ + c) * L_ + l] * att[b * C_ + c];
}

// ---------------------------------------------------------------------------
extern "C" void kernel_launch(void* const* d_in, const int* in_sizes, int n_in,
                              void* d_out, int out_size, void* d_ws, size_t ws_size,
                              hipStream_t stream)
{
    const float* input      = (const float*)d_in[0];
    const float* ln1_g      = (const float*)d_in[1];
    const float* ln1_b      = (const float*)d_in[2];
    const float* skip_scale = (const float*)d_in[3];
    const float* skip_scale2= (const float*)d_in[4];
    const float* ln2_g      = (const float*)d_in[5];
    const float* ln2_b      = (const float*)d_in[6];
    const float* in_proj_w  = (const float*)d_in[7];
    const float* conv2d_w   = (const float*)d_in[8];
    const float* conv2d_b   = (const float*)d_in[9];
    const float* x_proj_w   = (const float*)d_in[10];
    const float* dt_proj_w  = (const float*)d_in[11];
    const float* dt_proj_b  = (const float*)d_in[12];
    const float* A_logs     = (const float*)d_in[13];
    const float* Ds         = (const float*)d_in[14];
    const float* out_norm_g = (const float*)d_in[15];
    const float* out_norm_b = (const float*)d_in[16];
    const float* out_proj_w = (const float*)d_in[17];
    const float* f1_w = (const float*)d_in[18];
    const float* f1_b = (const float*)d_in[19];
    const float* fm_w = (const float*)d_in[20];
    const float* fm_b = (const float*)d_in[21];
    const float* f2_w = (const float*)d_in[22];
    const float* f2_b = (const float*)d_in[23];
    const float* mg_in_w  = (const float*)d_in[24];
    const float* mg_dw3_w = (const float*)d_in[25];
    const float* mg_dw5_w = (const float*)d_in[26];
    const float* mg_out_w = (const float*)d_in[27];
    const float* ca_w1 = (const float*)d_in[28];
    const float* ca_b1 = (const float*)d_in[29];
    const float* ca_w2 = (const float*)d_in[30];
    const float* ca_b2 = (const float*)d_in[31];
    float* outp = (float*)d_out;

    // ---- workspace layout (floats) ----
    float* Wp   = (float*)d_ws;
    float* XN   = Wp;                  // (B,L,96)        786432
    float* XZ   = XN   + 786432;       // (B,L,384)       3145728
    float* XC   = XZ   + 3145728;      // (B,192,L)       1572864
    float* XS   = XC   + 1572864;      // (B,4,192,L)     6291456
    float* XDBL = XS   + 6291456;      // (B,4,48,L) pad  1572864
    float* OY   = XDBL + 1572864;      // (B,4,192,L)     6291456
    float* YMUL = OY   + 6291456;      // (B,L,192)       1572864
    float* ATTN = YMUL + 1572864;      // (B,L,96)        786432
    float* XF   = ATTN + 786432;       // (B,48,L)        393216
    float* FWre = XF   + 393216;       // (B,48,64,33)    202752
    float* FWim = FWre + 202752;
    float* FR   = FWim + 202752;       // (B,96,64,33)    405504
    float* FRM  = FR   + 405504;
    float* IHre = FRM  + 405504;       // (B,48,64,33)
    float* IHim = IHre + 202752;
    float* XFI  = IHim + 202752;       // (B,48,64,64)    393216
    float* FFTO = XFI  + 393216;       // (B,L,96)        786432
    float* XRES = FFTO + 786432;       // (B,L,96)
    float* XN2  = XRES + 786432;       // (B,L,96)
    float* XPW  = XN2  + 786432;       // (4,48,192) padded x_proj_w  36864
    // MGCB aliases dead SS2D buffers:
    float* Pp   = XS;                  // (B,384,L)     3145728  (<= XS size)
    float* D1   = XS + 3145728;        // (B,288,L)     2359296
    float* X2D  = OY;                  // (B,96,L)      786432
    float* YM   = OY + 786432;         // (B,192,L)     1572864
    float* Y2   = OY + 786432 + 1572864; // (B,96,L)    786432
    float* POOL = Y2 + 786432;         // (B,96)
    float* ATT  = POOL + 192;          // (B,96)

    const int rows = B_ * L_;          // 8192

    // 0) pad x_proj_w to 48 rows per k
    k_pad_xpw<<<dim3((4 * 48 * DI_ + 255) / 256), dim3(256), 0, stream>>>(x_proj_w, XPW, 4 * 48 * DI_);

    // 1) xn = LN(input)
    k_ln96<<<dim3(rows / 8), dim3(256), 0, stream>>>(input, ln1_g, ln1_b, XN, rows, 1e-6f);

    // 2) xz = xn @ in_proj_w^T   [8192x96]x[96->384]
    k_gemm_f<true,true><<<dim3(rows / 16, 384 / 16, 1), dim3(32), 0, stream>>>(
        XN, in_proj_w, XZ, nullptr, 96,
        96LL, 1LL, 1LL, 96LL, 384LL, 1LL, 0LL, 0LL, 0LL, 0);

    // 3) xc = silu(dwconv3(xx) + b)  (xx = first 192 ch of xz, channels-last)
    {
        int total = B_ * DI_ * L_;
        k_dwconv<3><<<dim3(total / 256), dim3(256), 0, stream>>>(
            XZ, (long long)L_ * 384, 1LL, 384LL, conv2d_w, conv2d_b, XC, DI_, 1, total);
    }

    // 4) build 4-direction xs
    {
        int total = B_ * DI_ * L_;
        k_build_xs<<<dim3(total / 256), dim3(256), 0, stream>>>(XC, XS, total);
    }

    // 5) x_dbl[b,k] = xpw[k] @ xs[b,k]   [48x192]x[192x4096], batched over b
    for (int k = 0; k < K_; ++k) {
        k_gemm_f<true,false><<<dim3(3, L_ / 16, B_), dim3(32), 0, stream>>>(
            XPW + (long long)k * 48 * DI_,
            XS + (long long)k * DI_ * L_,
            XDBL + (long long)k * 48 * L_,
            nullptr, DI_,
            (long long)DI_, 1LL, (long long)L_, 1LL, (long long)L_, 1LL,
            0LL, (long long)K_ * DI_ * L_, (long long)K_ * 48 * L_, 0);
    }

    // 6) selective scan (dt-proj + softplus fused)
    k_scan<<<dim3(B_ * K_), dim3(DI_), 0, stream>>>(XS, XDBL, A_logs, dt_proj_w, dt_proj_b, Ds, OY);

    // 7) merge directions + out-norm LN + silu(z) gate
    k_merge<<<dim3(rows), dim3(DI_), 0, stream>>>(OY, XZ, out_norm_g, out_norm_b, YMUL);

    // 8) attn = ymul @ out_proj_w^T   [8192x192]x[192->96]
    k_gemm_f<true,true><<<dim3(rows / 16, 96 / 16, 1), dim3(32), 0, stream>>>(
        YMUL, out_proj_w, ATTN, nullptr, DI_,
        (long long)DI_, 1LL, 1LL, (long long)DI_, 96LL, 1LL, 0LL, 0LL, 0LL, 0);

    // 9) FFT branch: xf = silu(conv1x1(xn, f1)) -> channel planes (B,48,L)
    k_gemm_f<true,true><<<dim3(L_ / 16, 48 / 16, B_), dim3(32), 0, stream>>>(
        XN, f1_w, XF, f1_b, 96,
        96LL, 1LL, 1LL, 96LL, 1LL, (long long)L_,
        (long long)L_ * 96, 0LL, 48LL * L_, 2);

    // 10-11) forward rfft2 (ortho)
    {
        int total = B_ * 48 * 64 * 33;
        k_dft_r2c_w<<<dim3((total + 255) / 256), dim3(256), 0, stream>>>(XF, FWre, FWim, total);
        k_dft_h_fwd<<<dim3((total + 255) / 256), dim3(256), 0, stream>>>(FWre, FWim, FR, total);
    }

    // 12) freq conv1x1 (fm) + silu   M=2112 pixels, K=N=96
    k_gemm_f<false,true><<<dim3(2112 / 16, 96 / 16, B_), dim3(32), 0, stream>>>(
        FR, fm_w, FRM, fm_b, 96,
        1LL, 2112LL, 1LL, 96LL, 1LL, 2112LL,
        96LL * 2112, 0LL, 96LL * 2112, 2);

    // 13-14) inverse irfft2 (ortho)
    {
        int total = B_ * 48 * 64 * 33;
        k_dft_h_inv<<<dim3((total + 255) / 256), dim3(256), 0, stream>>>(FRM, IHre, IHim, total);
        int total2 = B_ * 48 * L_;
        k_idft_c2r_w<<<dim3(total2 / 256), dim3(256), 0, stream>>>(IHre, IHim, XFI, total2);
    }

    // 15) fft out conv1x1 (f2) + bias -> channels-last (B,L,96)
    k_gemm_f<false,true><<<dim3(L_ / 16, 96 / 16, B_), dim3(32), 0, stream>>>(
        XFI, f2_w, FFTO, f2_b, 48,
        1LL, (long long)L_, 1LL, 48LL, 96LL, 1LL,
        48LL * L_, 0LL, (long long)L_ * 96, 1);

    // 16) x = input*skip + attn + fft ; 17) xn2 = LN(x)
    {
        int total = B_ * L_ * C_;
        k_combine<<<dim3(total / 256), dim3(256), 0, stream>>>(input, skip_scale, ATTN, FFTO, XRES, total);
    }
    k_ln96<<<dim3(rows / 8), dim3(256), 0, stream>>>(XRES, ln2_g, ln2_b, XN2, rows, 1e-5f);

    // 18) p = conv1x1(xn2, mg_in) -> channel planes (B,384,L)
    k_gemm_f<true,true><<<dim3(L_ / 16, 384 / 16, B_), dim3(32), 0, stream>>>(
        XN2, mg_in_w, Pp, nullptr, 96,
        96LL, 1LL, 1LL, 96LL, 1LL, (long long)L_,
        (long long)L_ * 96, 0LL, 384LL * L_, 0);

    // 19) d1 = dwconv3(p[:, :288]);  20) x2d = dwconv5(p[:, 288:])
    {
        int total = B_ * 288 * L_;
        k_dwconv<3><<<dim3(total / 256), dim3(256), 0, stream>>>(
            Pp, 384LL * L_, (long long)L_, 1LL, mg_dw3_w, nullptr, D1, 288, 0, total);
        int total2 = B_ * 96 * L_;
        k_dwconv<5><<<dim3(total2 / 256), dim3(256), 0, stream>>>(
            Pp + 288LL * L_, 384LL * L_, (long long)L_, 1LL, mg_dw5_w, nullptr, X2D, 96, 0, total2);
    }

    // 21) y = gelu(x1_1) * concat(x1_2, x2d)
    {
        int total = B_ * DI_ * L_;
        k_gelu_mul<<<dim3(total / 256), dim3(256), 0, stream>>>(D1, X2D, YM, total);
    }

    // 22) y2 = conv1x1(y, mg_out) -> (B,96,L)
    k_gemm_f<false,true><<<dim3(L_ / 16, 96 / 16, B_), dim3(32), 0, stream>>>(
        YM, mg_out_w, Y2, nullptr, DI_,
        1LL, (long long)L_, 1LL, (long long)DI_, 1LL, (long long)L_,
        (long long)DI_ * L_, 0LL, 96LL * L_, 0);

    // 23-24) channel attention
    k_pool<<<dim3(B_ * 96), dim3(256), 0, stream>>>(Y2, POOL);
    k_ca<<<dim3(B_), dim3(96), 0, stream>>>(POOL, ca_w1, ca_b1, ca_w2, ca_b2, ATT);

    // 25) out = x*skip2 + y2*att (transposed to channels-last)
    {
        int total = B_ * L_ * C_;
        k_final<<<dim3(total / 256), dim3(256), 0, stream>>>(XRES, skip_scale2, Y2, ATT, outp, total);
    }
}